// Net_64682207478622
// MI455X (gfx1250) — compile-verified
//
#include <hip/hip_runtime.h>
#include <hip/hip_bf16.h>

// ---------------------------------------------------------------------------
// 2-layer GCN for MI455X (gfx1250, wave32).
//   h1   = x @ W1                       (WMMA f32 16x16x4, LDS-staged W1)
//   a1   = scatter(norm * h1[src]) + self + b1, ReLU
//   h2   = a1 @ W2                      (tiny 16->2, scalar)
//   a2   = scatter(norm * h2[src]) + self + b2
//   out  = log_softmax(a2)
// ---------------------------------------------------------------------------

typedef float v2f __attribute__((ext_vector_type(2)));
typedef float v8f __attribute__((ext_vector_type(8)));

#define FIN   512
#define FH    16
#define FOUT  2

// -------------------- degree / normalization --------------------

__global__ void k_init_deg(float* __restrict__ deg, int n) {
    int i = blockIdx.x * blockDim.x + threadIdx.x;
    if (i < n) deg[i] = 1.0f;                      // self loop counts once
}

__global__ void k_deg_accum(const int* __restrict__ ei, int E,
                            float* __restrict__ deg) {
    int e = blockIdx.x * blockDim.x + threadIdx.x;
    if (e < E) atomicAdd(&deg[ei[E + e]], 1.0f);   // dst row of edge_index
}

__global__ void k_rsqrt_inplace(float* __restrict__ buf, int n) {
    int i = blockIdx.x * blockDim.x + threadIdx.x;
    if (i < n) buf[i] = rsqrtf(buf[i]);            // deg >= 1 always
}

// -------------------- GEMM1: h = x @ W1 via V_WMMA_F32_16X16X4_F32 ---------
// One wave computes one 16x16 output tile (16 nodes x 16 features).
// A (16x4 f32) layout: lane = half*16 + m ; v[0]=K(2*half), v[1]=K(2*half+1)
// B (4x16 f32) layout: lane = half*16 + n ; v[0]=row(2*half), v[1]=row(2*half+1)
// C/D (16x16 f32): VGPR r -> row r + 8*half, col = lane&15

__global__ void __launch_bounds__(256)
k_gemm1_wmma(const float* __restrict__ x, const float* __restrict__ W1,
             float* __restrict__ h, int n_tiles) {
    __shared__ float lw[FIN * FH];                 // 32 KB: all of W1
    int tid = threadIdx.x;
    for (int i = tid; i < FIN * FH; i += 256) lw[i] = W1[i];
    __syncthreads();

    int wave = tid >> 5;
    int lane = tid & 31;
    int tile = blockIdx.x * 8 + wave;
    if (tile >= n_tiles) return;                   // wave-uniform: EXEC stays full

    int half = lane >> 4;                          // 0 or 1
    int mn   = lane & 15;                          // row (A) / col (B,C)
    int kofs = 2 * half;

    const float* __restrict__ xrow = x + (size_t)(tile * 16 + mn) * FIN;

    v8f c = {};
    #pragma unroll 4
    for (int k = 0; k < FIN; k += 4) {
        v2f a = *(const v2f*)(xrow + k + kofs);    // 8B aligned (k+kofs even)
        v2f b;
        b.x = lw[(k + kofs)     * FH + mn];
        b.y = lw[(k + kofs + 1) * FH + mn];
        c = __builtin_amdgcn_wmma_f32_16x16x4_f32(
                /*neg_a=*/false, a, /*neg_b=*/false, b,
                /*c_mod=*/(short)0, c, /*reuse_a=*/false, /*reuse_b=*/false);
    }

    float* __restrict__ hrow = h + (size_t)tile * 16 * FH;
    #pragma unroll
    for (int r = 0; r < 8; ++r)
        hrow[(r + 8 * half) * FH + mn] = c[r];
}

// -------------------- scatter layer 1 --------------------

__global__ void k_scatter1_init(const float* __restrict__ h,
                                const float* __restrict__ dinv,
                                float* __restrict__ agg1, int n) {
    int t = blockIdx.x * blockDim.x + threadIdx.x;     // n*16 threads
    if (t < n * FH) {
        int i = t >> 4;
        float w = dinv[i];
        agg1[t] = h[t] * w * w;                        // self-loop term
    }
}

__global__ void k_scatter1(const int* __restrict__ ei, int E,
                           const float* __restrict__ h,
                           const float* __restrict__ dinv,
                           float* __restrict__ agg1) {
    int t = blockIdx.x * blockDim.x + threadIdx.x;     // E*16 threads
    if (t < E * FH) {
        int e = t >> 4, f = t & 15;
        int s = ei[e], d = ei[E + e];
        float w = dinv[s] * dinv[d];
        atomicAdd(&agg1[d * FH + f], h[s * FH + f] * w);
    }
}

// -------------------- layer 2: h2 = relu(agg1 + b1) @ W2 --------------------

__global__ void k_layer2(const float* __restrict__ agg1,
                         const float* __restrict__ b1,
                         const float* __restrict__ W2,
                         float* __restrict__ h2, int n) {
    int i = blockIdx.x * blockDim.x + threadIdx.x;
    if (i < n) {
        float a0 = 0.0f, a1 = 0.0f;
        #pragma unroll
        for (int k = 0; k < FH; ++k) {
            float v = fmaxf(agg1[(size_t)i * FH + k] + b1[k], 0.0f);
            a0 = fmaf(v, W2[2 * k],     a0);
            a1 = fmaf(v, W2[2 * k + 1], a1);
        }
        h2[2 * i]     = a0;
        h2[2 * i + 1] = a1;
    }
}

// -------------------- scatter layer 2 --------------------

__global__ void k_scatter2_init(const float* __restrict__ h2,
                                const float* __restrict__ dinv,
                                float* __restrict__ agg2, int n) {
    int t = blockIdx.x * blockDim.x + threadIdx.x;     // n*2 threads
    if (t < n * FOUT) {
        int i = t >> 1;
        float w = dinv[i];
        agg2[t] = h2[t] * w * w;
    }
}

__global__ void k_scatter2(const int* __restrict__ ei, int E,
                           const float* __restrict__ h2,
                           const float* __restrict__ dinv,
                           float* __restrict__ agg2) {
    int t = blockIdx.x * blockDim.x + threadIdx.x;     // E*2 threads
    if (t < E * FOUT) {
        int e = t >> 1, f = t & 1;
        int s = ei[e], d = ei[E + e];
        float w = dinv[s] * dinv[d];
        atomicAdd(&agg2[d * FOUT + f], h2[s * FOUT + f] * w);
    }
}

// -------------------- bias + log_softmax --------------------

__global__ void k_logsoftmax(const float* __restrict__ agg2,
                             const float* __restrict__ b2,
                             float* __restrict__ out, int n) {
    int i = blockIdx.x * blockDim.x + threadIdx.x;
    if (i < n) {
        float a0 = agg2[2 * i]     + b2[0];
        float a1 = agg2[2 * i + 1] + b2[1];
        float m  = fmaxf(a0, a1);
        float ls = logf(expf(a0 - m) + expf(a1 - m));
        out[2 * i]     = a0 - m - ls;
        out[2 * i + 1] = a1 - m - ls;
    }
}

// -------------------- launch --------------------

extern "C" void kernel_launch(void* const* d_in, const int* in_sizes, int n_in,
                              void* d_out, int out_size, void* d_ws, size_t ws_size,
                              hipStream_t stream) {
    const float* x  = (const float*)d_in[0];   // [N, 512]
    const int*   ei = (const int*)d_in[1];     // [2, E]
    const float* W1 = (const float*)d_in[2];   // [512, 16]
    const float* b1 = (const float*)d_in[3];   // [16]
    const float* W2 = (const float*)d_in[4];   // [16, 2]
    const float* b2 = (const float*)d_in[5];   // [2]
    float* out      = (float*)d_out;           // [N, 2] log-probs

    const int N = in_sizes[0] / FIN;           // 100000
    const int E = in_sizes[1] / 2;             // 3200000

    // workspace layout (floats)
    float* ws   = (float*)d_ws;
    float* dinv = ws;                          // N        (deg -> dinv in place)
    float* h    = dinv + N;                    // N*16
    float* agg1 = h    + (size_t)N * FH;       // N*16
    float* h2   = agg1 + (size_t)N * FH;       // N*2
    float* agg2 = h2   + (size_t)N * FOUT;     // N*2

    const int B = 256;
    auto blk = [](long long t, int b) { return (int)((t + b - 1) / b); };

    // normalization
    k_init_deg   <<<blk(N, B), B, 0, stream>>>(dinv, N);
    k_deg_accum  <<<blk(E, B), B, 0, stream>>>(ei, E, dinv);
    k_rsqrt_inplace<<<blk(N, B), B, 0, stream>>>(dinv, N);

    // layer 1
    const int n_tiles = N / 16;                // 6250, exact
    k_gemm1_wmma <<<blk(n_tiles, 8), B, 0, stream>>>(x, W1, h, n_tiles);
    k_scatter1_init<<<blk((long long)N * FH, B), B, 0, stream>>>(h, dinv, agg1, N);
    k_scatter1   <<<blk((long long)E * FH, B), B, 0, stream>>>(ei, E, h, dinv, agg1);

    // layer 2
    k_layer2     <<<blk(N, B), B, 0, stream>>>(agg1, b1, W2, h2, N);
    k_scatter2_init<<<blk((long long)N * FOUT, B), B, 0, stream>>>(h2, dinv, agg2, N);
    k_scatter2   <<<blk((long long)E * FOUT, B), B, 0, stream>>>(ei, E, h2, dinv, agg2);

    // output
    k_logsoftmax <<<blk(N, B), B, 0, stream>>>(agg2, b2, out, N);
}